// SuperLinearTransformerBlock_4501125726458
// MI455X (gfx1250) — compile-verified
//
#include <hip/hip_runtime.h>

// KDA chunked gated delta-rule scan for MI455X (gfx1250, wave32).
// - one workgroup per (b,h); S state resident in LDS across the 64-chunk scan
// - all matmuls: V_WMMA_F32_16X16X4_F32 (fp32 in, fp32 accum)
// - chunk input staged by the Tensor Data Mover (tensor_load_to_lds) with
//   LDS row padding folded into the descriptor (pad_interval/pad_amount)
// - register blocking: each wave owns one 16-row block and sweeps N-tiles,
//   so A fragments are loaded once per k-step (cuts DS traffic ~2x)

typedef float v2f __attribute__((ext_vector_type(2)));
typedef float v8f __attribute__((ext_vector_type(8)));
typedef unsigned int ui32x4 __attribute__((ext_vector_type(4)));
typedef int i32x8 __attribute__((ext_vector_type(8)));
typedef int i32x4 __attribute__((ext_vector_type(4)));

#define B_N   2
#define T_N   4096
#define H_N   16
#define K_D   128
#define V_D   128
#define C_N   64
#define NC_N  64
#define LDA   140   // 128 + 12: bank-conflict-free padded stride (64 banks)
#define LDB   76    // 64 + 12

// LDS layout (in floats)
#define S_OFF  0
#define QT_OFF (S_OFF + K_D * LDA)        // 17920
#define UB_OFF (QT_OFF + C_N * LDA)       // +8960
#define WB_OFF (UB_OFF + C_N * LDA)
#define VB_OFF (WB_OFF + C_N * LDA)
#define AB_OFF (VB_OFF + C_N * LDA)       // 64x64 (A = -L, Neumann P ping)
#define IB_OFF (AB_OFF + C_N * LDB)
#define T0_OFF (IB_OFF + C_N * LDB)
#define T1_OFF (T0_OFF + C_N * LDB)
#define P1_OFF (T1_OFF + C_N * LDB)
#define WL_OFF (P1_OFF + C_N * LDB)       // 128
#define QI_OFF (WL_OFF + K_D)             // 64
#define KI_OFF (QI_OFF + C_N)             // 64
#define BE_OFF (KI_OFF + C_N)             // 64
#define SM_TOT (BE_OFF + C_N)             // 78400 floats = 313600 B

__device__ __forceinline__ v8f wmma4(v2f a, v2f b, v8f c) {
  return __builtin_amdgcn_wmma_f32_16x16x4_f32(
      false, a, false, b, (short)0, c, false, false);
}

// NT tiles along N sharing one A fragment per k-step.
// TA: element (m,k) at A[k*lda + m].
template <bool TA, int NT>
__device__ __forceinline__ void mm_multi(const float* __restrict__ A, int lda,
                                         const float* __restrict__ B, int ldb,
                                         int m0, int n0, int kLen, v8f* acc) {
  const int lane = threadIdx.x & 31;
  const int r    = lane & 15;
  const int koff = (lane >> 4) << 1;
  for (int k = 0; k < kLen; k += 4) {
    const int ka = k + koff;
    v2f a;
    if (TA) { a.x = A[ka * lda + m0 + r];   a.y = A[(ka + 1) * lda + m0 + r]; }
    else    { a.x = A[(m0 + r) * lda + ka]; a.y = A[(m0 + r) * lda + ka + 1]; }
#pragma unroll
    for (int t = 0; t < NT; ++t) {
      const int nb = n0 + 16 * t;
      v2f b;
      b.x = B[ka * ldb + nb + r];
      b.y = B[(ka + 1) * ldb + nb + r];
      acc[t] = wmma4(a, b, acc[t]);
    }
  }
}

// Fused: acc0 += A0(m-block) * Bt^T, acc1 += A1(m-block) * Bt^T,
// sharing the B fragment (Bt element (k,n) at Bt[n*ldb + k]); 2 N-tiles.
__device__ __forceinline__ void mm_dualA_Bt(const float* __restrict__ A0,
                                            const float* __restrict__ A1,
                                            int lda,
                                            const float* __restrict__ Bt,
                                            int ldb, int m0, int n0, int kLen,
                                            v8f* acc0, v8f* acc1) {
  const int lane = threadIdx.x & 31;
  const int r    = lane & 15;
  const int koff = (lane >> 4) << 1;
  for (int k = 0; k < kLen; k += 4) {
    const int ka = k + koff;
    v2f a0, a1;
    a0.x = A0[(m0 + r) * lda + ka]; a0.y = A0[(m0 + r) * lda + ka + 1];
    a1.x = A1[(m0 + r) * lda + ka]; a1.y = A1[(m0 + r) * lda + ka + 1];
#pragma unroll
    for (int t = 0; t < 2; ++t) {
      const int nb = n0 + 16 * t;
      v2f b;
      b.x = Bt[(nb + r) * ldb + ka];
      b.y = Bt[(nb + r) * ldb + ka + 1];
      acc0[t] = wmma4(a0, b, acc0[t]);
      acc1[t] = wmma4(a1, b, acc1[t]);
    }
  }
}

// Neumann step: accT[t] += P(m-blk)*T(:,n), accP[t] += P(m-blk)*P(:,n),
// sharing the A fragment (rows of P); 2 N-tiles of each.
__device__ __forceinline__ void inv_step(const float* __restrict__ Pc,
                                         const float* __restrict__ Tc,
                                         int m0, int n0, v8f* accT, v8f* accP) {
  const int lane = threadIdx.x & 31;
  const int r    = lane & 15;
  const int koff = (lane >> 4) << 1;
  for (int k = 0; k < C_N; k += 4) {
    const int ka = k + koff;
    v2f a;
    a.x = Pc[(m0 + r) * LDB + ka]; a.y = Pc[(m0 + r) * LDB + ka + 1];
#pragma unroll
    for (int t = 0; t < 2; ++t) {
      const int nb = n0 + 16 * t;
      v2f bT, bP;
      bT.x = Tc[ka * LDB + nb + r]; bT.y = Tc[(ka + 1) * LDB + nb + r];
      bP.x = Pc[ka * LDB + nb + r]; bP.y = Pc[(ka + 1) * LDB + nb + r];
      accT[t] = wmma4(a, bT, accT[t]);
      accP[t] = wmma4(a, bP, accP[t]);
    }
  }
}

__device__ __forceinline__ v8f load_tile(const float* __restrict__ P, int ld,
                                         int m0, int n0) {
  const int lane = threadIdx.x & 31;
  const int nn = n0 + (lane & 15);
  const int mb = m0 + ((lane >> 4) << 3);
  v8f c;
#pragma unroll
  for (int r = 0; r < 8; ++r) c[r] = P[(mb + r) * ld + nn];
  return c;
}

__device__ __forceinline__ void store_tile(float* __restrict__ P, int ld,
                                           int m0, int n0, v8f c) {
  const int lane = threadIdx.x & 31;
  const int nn = n0 + (lane & 15);
  const int mb = m0 + ((lane >> 4) << 3);
#pragma unroll
  for (int r = 0; r < 8; ++r) P[(mb + r) * ld + nn] = c[r];
}

// TDM: 2D tile load, 64 rows x 128 dwords, row stride 2048 dwords in memory,
// LDS rows padded to 140 dwords via pad_interval=6 (128 dw) / pad_amount=11
// (12 dw).  6-arg builtin form (clang-23 / therock-10.0 headers).
__device__ __forceinline__ void tdm_load_2d(unsigned long long gaddr,
                                            unsigned int lds_byte) {
  ui32x4 g0;
  g0[0] = 1u;                                        // count=1, normal mode
  g0[1] = lds_byte;                                  // lds_addr (bytes)
  g0[2] = (unsigned int)(gaddr & 0xFFFFFFFFull);     // global_addr[31:0]
  g0[3] = (unsigned int)((gaddr >> 32) & 0x1FFFFFFull) | (2u << 30);  // type=2
  i32x8 g1;
  g1[0] = (int)((2u << 16) | (1u << 20) | (6u << 22) | (11u << 25));
  //            data_size=4B  pad_en     pad_int=128dw  pad_amt=12dw
  g1[1] = (int)((128u & 0xFFFFu) << 16);             // tensor_dim0 = 128
  g1[2] = (int)((64u & 0xFFFFu) << 16);              // tensor_dim1 = 64
  g1[3] = (int)(128u << 16);                         // tile_dim0 = 128
  g1[4] = (int)64u;                                  // tile_dim1 = 64
  g1[5] = (int)2048u;                                // tensor_dim0_stride
  g1[6] = 0;
  g1[7] = 0;
  i32x4 gz4 = {0, 0, 0, 0};
  i32x8 gz8 = {0, 0, 0, 0, 0, 0, 0, 0};
  __builtin_amdgcn_tensor_load_to_lds(g0, g1, gz4, gz4, gz8, 0);
}

__global__ __launch_bounds__(256) void kda_scan_kernel(
    const float* __restrict__ q, const float* __restrict__ k,
    const float* __restrict__ v, const float* __restrict__ lg,
    const float* __restrict__ beta, float* __restrict__ out) {
  extern __shared__ float sm[];
  float* Sst = sm + S_OFF;
  float* qt  = sm + QT_OFF;
  float* ub  = sm + UB_OFF;
  float* wb  = sm + WB_OFF;
  float* vb  = sm + VB_OFF;
  float* Ab  = sm + AB_OFF;
  float* Ib  = sm + IB_OFF;
  float* T0  = sm + T0_OFF;
  float* T1  = sm + T1_OFF;
  float* P1  = sm + P1_OFF;
  float* wlast = sm + WL_OFF;
  float* qinv  = sm + QI_OFF;
  float* kinv  = sm + KI_OFF;
  float* bet   = sm + BE_OFF;

  const int tid  = threadIdx.x;
  const int wave = tid >> 5;
  const int lane = tid & 31;
  const int b = blockIdx.x / H_N;
  const int h = blockIdx.x % H_N;
  const float scale = 0.08838834764831845f;  // 128^-0.5

  const unsigned long long qg = (unsigned long long)(uintptr_t)q;
  const unsigned long long kg = (unsigned long long)(uintptr_t)k;
  const unsigned long long vg = (unsigned long long)(uintptr_t)v;
  const unsigned long long gg = (unsigned long long)(uintptr_t)lg;

  for (int i = tid; i < K_D * LDA; i += 256) Sst[i] = 0.0f;

  for (int n = 0; n < NC_N; ++n) {
    __syncthreads();
    // ---- stage chunk via Tensor Data Mover (wave 0 issues, all consume) ----
    if (wave == 0) {
      const unsigned long long off =
          (((unsigned long long)b * T_N + (unsigned long long)n * C_N) * H_N +
           (unsigned long long)h) * (unsigned long long)(K_D * 4);
      tdm_load_2d(qg + off, QT_OFF * 4u);
      tdm_load_2d(kg + off, UB_OFF * 4u);
      tdm_load_2d(vg + off, VB_OFF * 4u);
      tdm_load_2d(gg + off, WB_OFF * 4u);
    }
    if (tid < C_N) {
      const long t = (long)n * C_N + tid;
      bet[tid] = beta[((long)b * T_N + t) * H_N + h];
    }
    if (wave == 0) __builtin_amdgcn_s_wait_tensorcnt(0);
    __syncthreads();

    // ---- l2-norm factors per row ----
    if (tid < C_N) {
      float sq = 0.0f, sk = 0.0f;
      for (int kk = 0; kk < K_D; ++kk) {
        const float a = qt[tid * LDA + kk]; sq += a * a;
        const float c2 = ub[tid * LDA + kk]; sk += c2 * c2;
      }
      qinv[tid] = 1.0f / fmaxf(sqrtf(sq), 1e-12f);
      kinv[tid] = 1.0f / fmaxf(sqrtf(sk), 1e-12f);
    }
    // ---- cumsum of log_g over time per channel, clip at -15 ----
    if (tid < K_D) {
      float acc = 0.0f;
      for (int c = 0; c < C_N; ++c) {
        acc += wb[c * LDA + tid];
        wb[c * LDA + tid] = fmaxf(acc, -15.0f);
      }
      wlast[tid] = expf(fmaxf(acc, -15.0f));
    }
    __syncthreads();
    // ---- elementwise gating ----
    for (int e = tid; e < C_N * K_D; e += 256) {
      const int c = e >> 7, kk = e & 127;
      const int idx = c * LDA + kk;
      const float cum = wb[idx];
      const float W = expf(cum), Wi = expf(-cum);
      const float kn = ub[idx] * kinv[c];
      const float qn = qt[idx] * qinv[c] * scale;
      const float be = bet[c];
      ub[idx] = kn * Wi;
      qt[idx] = qn * W;
      wb[idx] = be * kn * W;
      vb[idx] *= be;
    }
    __syncthreads();

    // ---- A = -strict_tril(beta_w @ u^T), intra = tril(qt @ u^T) ----
    {
      const int m0 = (wave >> 1) * 16, n0 = (wave & 1) * 32;
      v8f za[2] = {}, zi[2] = {};
      if (m0 >= n0)  // wave-uniform; fully-upper tile pairs are all zero
        mm_dualA_Bt(wb, qt, LDA, ub, LDA, m0, n0, K_D, za, zi);
      const int mb = m0 + ((lane >> 4) << 3);
#pragma unroll
      for (int t = 0; t < 2; ++t) {
        const int nn = n0 + 16 * t + (lane & 15);
#pragma unroll
        for (int r = 0; r < 8; ++r) {
          const int M = mb + r;
          Ab[M * LDB + nn] = (M > nn) ? -za[t][r] : 0.0f;
          Ib[M * LDB + nn] = (M >= nn) ? zi[t][r] : 0.0f;
        }
      }
    }
    __syncthreads();

    // ---- (I+L)^-1 via Neumann doubling: T=I, P=A; 6x {T+=P*T; P=P*P} ----
    for (int e = tid; e < C_N * C_N; e += 256)
      T0[(e >> 6) * LDB + (e & 63)] = ((e >> 6) == (e & 63)) ? 1.0f : 0.0f;
    __syncthreads();
    float *Tc = T0, *Tx = T1, *Pc = Ab, *Px = P1;
    {
      const int m0 = (wave >> 1) * 16, n0 = (wave & 1) * 32;
      for (int it = 0; it < 6; ++it) {
        v8f accT[2], accP[2] = {};
        accT[0] = load_tile(Tc, LDB, m0, n0);
        accT[1] = load_tile(Tc, LDB, m0, n0 + 16);
        inv_step(Pc, Tc, m0, n0, accT, accP);
        store_tile(Tx, LDB, m0, n0, accT[0]);
        store_tile(Tx, LDB, m0, n0 + 16, accT[1]);
        store_tile(Px, LDB, m0, n0, accP[0]);
        store_tile(Px, LDB, m0, n0 + 16, accP[1]);
        __syncthreads();
        float* t2 = Tc; Tc = Tx; Tx = t2;
        t2 = Pc; Pc = Px; Px = t2;
      }
    }

    const int m0e = (wave >> 1) * 16;   // each wave owns one 16-row block
    const int n0e = (wave & 1) * 64;    // and a 64-col half (4 N-tiles)
    v8f zero = {};

    // ---- effective_v = T @ beta_v (in place in vb) ----
    {
      v8f acc[4] = {zero, zero, zero, zero};
      mm_multi<false, 4>(Tc, LDB, vb, LDA, m0e, n0e, C_N, acc);
      __syncthreads();  // all reads of beta_v complete
#pragma unroll
      for (int t = 0; t < 4; ++t)
        store_tile(vb, LDA, m0e, n0e + 16 * t, acc[t]);
    }
    // ---- effective_w = T @ beta_w, stored negated (in place in wb) ----
    {
      v8f acc[4] = {zero, zero, zero, zero};
      mm_multi<false, 4>(Tc, LDB, wb, LDA, m0e, n0e, C_N, acc);
      __syncthreads();  // all reads of beta_w complete
#pragma unroll
      for (int t = 0; t < 4; ++t) {
#pragma unroll
        for (int r = 0; r < 8; ++r) acc[t][r] = -acc[t][r];
        store_tile(wb, LDA, m0e, n0e + 16 * t, acc[t]);
      }
    }
    __syncthreads();  // -effw visible everywhere

    // ---- delta = effv + (-effw) @ S (in place in vb; C-init is own tile) ----
    {
      v8f acc[4];
#pragma unroll
      for (int t = 0; t < 4; ++t) acc[t] = load_tile(vb, LDA, m0e, n0e + 16 * t);
      mm_multi<false, 4>(wb, LDA, Sst, LDA, m0e, n0e, K_D, acc);
#pragma unroll
      for (int t = 0; t < 4; ++t) store_tile(vb, LDA, m0e, n0e + 16 * t, acc[t]);
    }
    __syncthreads();  // delta visible everywhere

    // ---- o = qt @ S + intra @ delta -> global ----
    {
      v8f acc[4] = {zero, zero, zero, zero};
      mm_multi<false, 4>(qt, LDA, Sst, LDA, m0e, n0e, K_D, acc);
      mm_multi<false, 4>(Ib, LDB, vb, LDA, m0e, n0e, C_N, acc);
      const int mb = m0e + ((lane >> 4) << 3);
#pragma unroll
      for (int t = 0; t < 4; ++t) {
        const int nn = n0e + 16 * t + (lane & 15);
#pragma unroll
        for (int r = 0; r < 8; ++r) {
          const long tt = (long)n * C_N + mb + r;
          out[(((long)b * T_N + tt) * H_N + h) * V_D + nn] = acc[t][r];
        }
      }
    }
    __syncthreads();  // all reads of old S complete

    // ---- S = diag(w_last) * (S + u^T @ delta), in place ----
    {
      const int m0s = wave * 16;
      const int mb = m0s + ((lane >> 4) << 3);
#pragma unroll
      for (int half = 0; half < 2; ++half) {
        const int n0s = half * 64;
        v8f acc[4];
#pragma unroll
        for (int t = 0; t < 4; ++t)
          acc[t] = load_tile(Sst, LDA, m0s, n0s + 16 * t);
        mm_multi<true, 4>(ub, LDA, vb, LDA, m0s, n0s, C_N, acc);
#pragma unroll
        for (int t = 0; t < 4; ++t) {
#pragma unroll
          for (int r = 0; r < 8; ++r) acc[t][r] *= wlast[mb + r];
          store_tile(Sst, LDA, m0s, n0s + 16 * t, acc[t]);
        }
      }
    }
  }
}

extern "C" void kernel_launch(void* const* d_in, const int* in_sizes, int n_in,
                              void* d_out, int out_size, void* d_ws, size_t ws_size,
                              hipStream_t stream) {
  (void)in_sizes; (void)n_in; (void)out_size; (void)d_ws; (void)ws_size;
  const float* q    = (const float*)d_in[0];
  const float* k    = (const float*)d_in[1];
  const float* v    = (const float*)d_in[2];
  const float* lg   = (const float*)d_in[3];
  const float* beta = (const float*)d_in[4];
  float* out = (float*)d_out;

  const size_t shmem = (size_t)SM_TOT * sizeof(float);  // 313600 B < 320 KB
  (void)hipFuncSetAttribute((const void*)kda_scan_kernel,
                            hipFuncAttributeMaxDynamicSharedMemorySize,
                            (int)shmem);
  kda_scan_kernel<<<dim3(B_N * H_N), dim3(256), shmem, stream>>>(q, k, v, lg,
                                                                 beta, out);
}